// MahalanobisLoss_54941221650666
// MI455X (gfx1250) — compile-verified
//
#include <hip/hip_runtime.h>
#include <stdint.h>

#define TILE          256                 // elements per tile
#define TILE_PR_BYTES 3072                // 256*3*4
#define TILE_CH_BYTES 6144                // 256*6*4
#define TILE_BYTES    12288               // 256*12*4
#define NTHREADS      256
#define MAXGRID       2048

// ---- CDNA5 async global->LDS (TDM-adjacent path, tracked by ASYNCcnt) ----
__device__ __forceinline__ void async_b128(uint32_t ldsOff, const void* g) {
    asm volatile("global_load_async_to_lds_b128 %0, %1, off"
                 :
                 : "v"(ldsOff), "v"((unsigned long long)(uintptr_t)g)
                 : "memory");
}

// Per-element loss: e^T (L L^T)^-1 e + log(sum(LL^T)+1)
// via forward substitution (L y = e -> md = |y|^2) and
// sum(Q) = (a+b+d)^2 + (c+e)^2 + f^2.
__device__ __forceinline__ float elem_loss(float p0, float p1, float p2,
                                           float q0, float q1, float q2,
                                           float a, float b, float c,
                                           float d, float e, float f) {
    float e0 = p0 - q0, e1 = p1 - q1, e2 = p2 - q2;
    float s1 = a + b + d, s2 = c + e;
    float normQ = s1 * s1 + s2 * s2 + f * f;
    float y0 = e0 / a;
    float y1 = (e1 - b * y0) / c;
    float y2 = (e2 - d * y0 - e * y1) / f;
    return y0 * y0 + y1 * y1 + y2 * y2 + logf(normQ + 1.0f);
}

// Issue exactly 3 async b128 loads per thread (branch-free source select so
// every wave's ASYNCcnt advances by exactly 3 per tile -> precise waits).
// LDS layout per buffer: [pr 3072B | x 3072B | chol 6144B], chunk c -> lds + c*16.
__device__ __forceinline__ void issue_tile(const float* pr, const float* x,
                                           const float* ch, long tile,
                                           uint32_t bufByte, int t) {
    const char* prB = (const char*)pr + tile * (long)TILE_PR_BYTES;
    const char* xB  = (const char*)x  + tile * (long)TILE_PR_BYTES;
    const char* chB = (const char*)ch + tile * (long)TILE_CH_BYTES;
    // chunk t in [0,256): pr for c<192, else x[c-192]
    const char* s0 = (t < 192) ? (prB + t * 16) : (xB + (t - 192) * 16);
    async_b128(bufByte + (uint32_t)t * 16u, s0);
    // chunk 256+t: x[64+t] for t<128, else chol[t-128]
    const char* s1 = (t < 128) ? (xB + (64 + t) * 16) : (chB + (t - 128) * 16);
    async_b128(bufByte + 4096u + (uint32_t)t * 16u, s1);
    // chunk 512+t: chol[128+t]
    async_b128(bufByte + 8192u + (uint32_t)t * 16u, chB + (128 + t) * 16);
}

__global__ __launch_bounds__(NTHREADS)
void mahal_main_kernel(const float* __restrict__ pr, const float* __restrict__ x,
                       const float* __restrict__ ch, float* __restrict__ partials,
                       long ntiles) {
    __shared__ __align__(16) float smem[2 * (TILE_BYTES / 4)];
    __shared__ float wsum[NTHREADS / 32];
    const uint32_t base = (uint32_t)(uintptr_t)&smem[0];  // flat low 32 bits == LDS byte offset
    const int t = threadIdx.x;
    const long stride = gridDim.x;
    long tile = blockIdx.x;
    int buf = 0;

    issue_tile(pr, x, ch, tile, base, t);  // prologue: ASYNCcnt = 3

    float acc = 0.0f;
    for (; tile < ntiles; tile += stride) {
        long nxt = tile + stride;
        if (nxt < ntiles) {
            issue_tile(pr, x, ch, nxt, base + (uint32_t)(buf ^ 1) * TILE_BYTES, t);
            asm volatile("s_wait_asynccnt 3" ::: "memory");  // oldest 3 (this tile) done
        } else {
            asm volatile("s_wait_asynccnt 0" ::: "memory");
        }
        __syncthreads();  // all waves' data landed in LDS

        const float* f = &smem[buf * (TILE_BYTES / 4)];
        int t3 = 3 * t, t6 = 6 * t;  // strides 3 & 6: bank-conflict-free (3t, 6t distinct mod 64 across 32 lanes)
        acc += elem_loss(f[t3], f[t3 + 1], f[t3 + 2],
                         f[768 + t3], f[768 + t3 + 1], f[768 + t3 + 2],
                         f[1536 + t6], f[1536 + t6 + 1], f[1536 + t6 + 2],
                         f[1536 + t6 + 3], f[1536 + t6 + 4], f[1536 + t6 + 5]);

        __syncthreads();  // everyone done reading buf before it's overwritten
        buf ^= 1;
    }

    // wave32 reduction, then cross-wave via LDS (deterministic order)
    for (int o = 16; o > 0; o >>= 1) acc += __shfl_down(acc, o, 32);
    if ((t & 31) == 0) wsum[t >> 5] = acc;
    __syncthreads();
    if (t == 0) {
        float s = 0.0f;
        for (int i = 0; i < NTHREADS / 32; ++i) s += wsum[i];
        partials[blockIdx.x] = s;
    }
}

__global__ __launch_bounds__(NTHREADS)
void mahal_rem_kernel(const float* __restrict__ pr, const float* __restrict__ x,
                      const float* __restrict__ ch, float* __restrict__ partial,
                      long start, long N) {
    __shared__ float wsum[NTHREADS / 32];
    float acc = 0.0f;
    for (long e = start + threadIdx.x; e < N; e += NTHREADS) {
        const float* P = pr + 3 * e;
        const float* X = x + 3 * e;
        const float* C = ch + 6 * e;
        acc += elem_loss(P[0], P[1], P[2], X[0], X[1], X[2],
                         C[0], C[1], C[2], C[3], C[4], C[5]);
    }
    for (int o = 16; o > 0; o >>= 1) acc += __shfl_down(acc, o, 32);
    if ((threadIdx.x & 31) == 0) wsum[threadIdx.x >> 5] = acc;
    __syncthreads();
    if (threadIdx.x == 0) {
        float s = 0.0f;
        for (int i = 0; i < NTHREADS / 32; ++i) s += wsum[i];
        partial[0] = s;
    }
}

__global__ __launch_bounds__(NTHREADS)
void mahal_finalize_kernel(const float* __restrict__ partials, int nparts,
                           float invN, float* __restrict__ out) {
    __shared__ float wsum[NTHREADS / 32];
    float acc = 0.0f;
    for (int i = threadIdx.x; i < nparts; i += NTHREADS) acc += partials[i];
    for (int o = 16; o > 0; o >>= 1) acc += __shfl_down(acc, o, 32);
    if ((threadIdx.x & 31) == 0) wsum[threadIdx.x >> 5] = acc;
    __syncthreads();
    if (threadIdx.x == 0) {
        float s = 0.0f;
        for (int i = 0; i < NTHREADS / 32; ++i) s += wsum[i];
        out[0] = s * invN;
    }
}

extern "C" void kernel_launch(void* const* d_in, const int* in_sizes, int n_in,
                              void* d_out, int out_size, void* d_ws, size_t ws_size,
                              hipStream_t stream) {
    const float* pr = (const float*)d_in[0];
    const float* x  = (const float*)d_in[1];
    const float* ch = (const float*)d_in[2];
    long N = (long)in_sizes[0] / 3;           // B*T elements
    long ntiles = N / TILE;
    long rem = N - ntiles * TILE;
    float* partials = (float*)d_ws;

    int grid = (int)(ntiles < (long)MAXGRID ? ntiles : (long)MAXGRID);
    int nparts = 0;
    if (grid > 0) {
        mahal_main_kernel<<<dim3(grid), dim3(NTHREADS), 0, stream>>>(
            pr, x, ch, partials, ntiles);
        nparts = grid;
    }
    if (rem > 0) {
        mahal_rem_kernel<<<dim3(1), dim3(NTHREADS), 0, stream>>>(
            pr, x, ch, partials + nparts, ntiles * (long)TILE, N);
        nparts += 1;
    }
    float invN = (N > 0) ? (float)(1.0 / (double)N) : 0.0f;
    mahal_finalize_kernel<<<dim3(1), dim3(NTHREADS), 0, stream>>>(
        partials, nparts, invN, (float*)d_out);
}